// GATLayer_76536317215335
// MI455X (gfx1250) — compile-verified
//
#include <hip/hip_runtime.h>
#include <math.h>

typedef float v2f __attribute__((ext_vector_type(2)));
typedef float v8f __attribute__((ext_vector_type(8)));

#define DIN   512
#define NH    8
#define DP    64
#define DOUT  512   // NH * DP

// ---------------------------------------------------------------------------
// Kernel 1: Wx = x @ W + b   (all 8 heads = one 512-wide GEMM)
// Block = 256 threads = 8 waves computing a 32(M) x 512(N) tile.
// Wave h owns head h's 64 columns with two 16-row accumulator sets, so each
// B fragment feeds 2 x v_wmma_f32_16x16x4_f32. A (32x32 chunk) staged in LDS.
// ---------------------------------------------------------------------------
__global__ __launch_bounds__(256) void gat_gemm(const float* __restrict__ x,
                                                const float* __restrict__ Ws,
                                                const float* __restrict__ Wb,
                                                float* __restrict__ Wx,
                                                int n_nodes)
{
    __shared__ float As[32 * 36];      // stride 36: 16B-aligned rows, conflict-free
    const int m0   = blockIdx.x << 5;
    const int head = threadIdx.x >> 5;
    const int lane = threadIdx.x & 31;
    const int hl   = lane >> 4;        // half-wave select (K pairs 0/1 vs 2/3)
    const int r    = lane & 15;

    const float* Wh = Ws + (size_t)head * (DIN * DP);   // [512][64] for this head

    v8f acc[2][4];
#pragma unroll
    for (int t = 0; t < 4; ++t) {
        float b = Wb[head * DP + t * 16 + r];
        acc[0][t] = (v8f){b, b, b, b, b, b, b, b};
        acc[1][t] = acc[0][t];
    }

    for (int k0 = 0; k0 < DIN; k0 += 32) {
        // cooperative stage of 32x32 A chunk (1024 floats, 4 per thread)
        {
            int i   = threadIdx.x * 4;
            int row = i >> 5;          // 0..31
            int kk  = i & 31;          // multiple of 4
            int grow = m0 + row;
            if (grow >= n_nodes) grow = n_nodes - 1;   // clamp, stores are guarded
            const float* src = x + (size_t)grow * DIN + k0 + kk;
            float4 v = *(const float4*)src;
            float* dst = &As[row * 36 + kk];
            dst[0] = v.x; dst[1] = v.y; dst[2] = v.z; dst[3] = v.w;
            if (k0 + 32 < DIN)
                __builtin_prefetch(src + 32, 0, 1);    // global_prefetch next chunk
        }
        __syncthreads();

#pragma unroll
        for (int kk = 0; kk < 32; kk += 4) {
            const int ka = kk + 2 * hl;                // this half-wave's K pair
            v2f a0, a1;
            a0.x = As[r * 36 + ka];        a0.y = As[r * 36 + ka + 1];
            a1.x = As[(r + 16) * 36 + ka]; a1.y = As[(r + 16) * 36 + ka + 1];
            const float* bp = Wh + (size_t)(k0 + ka) * DP + r;
#pragma unroll
            for (int t = 0; t < 4; ++t) {
                v2f b;
                b.x = bp[t * 16];          // B[ka][n]
                b.y = bp[t * 16 + DP];     // B[ka+1][n]
                acc[0][t] = __builtin_amdgcn_wmma_f32_16x16x4_f32(
                    false, a0, false, b, (short)0, acc[0][t], false, false);
                acc[1][t] = __builtin_amdgcn_wmma_f32_16x16x4_f32(
                    false, a1, false, b, (short)0, acc[1][t], false, false);
            }
        }
        __syncthreads();
    }

    // store: C/D layout -> VGPR v: row base+v (lanes 0-15) / base+v+8 (lanes 16-31)
#pragma unroll
    for (int u = 0; u < 2; ++u) {
#pragma unroll
        for (int t = 0; t < 4; ++t) {
#pragma unroll
            for (int v = 0; v < 8; ++v) {
                int row = m0 + u * 16 + v + 8 * hl;
                if (row < n_nodes) {
                    int col = head * DP + t * 16 + r;
                    Wx[(size_t)row * DOUT + col] = acc[u][t][v];
                }
            }
        }
    }
}

// ---------------------------------------------------------------------------
// Kernel 2: per-(node,head) attention coefficients
//   er[n,h] = Wx[n,h,:] . a_w[h, :64]    (target coefficient)
//   el[n,h] = Wx[n,h,:] . a_w[h, 64:]    (source coefficient)
// ---------------------------------------------------------------------------
__global__ __launch_bounds__(256) void gat_node_coef(const float* __restrict__ Wx,
                                                     const float* __restrict__ aw,
                                                     float* __restrict__ el,
                                                     float* __restrict__ er,
                                                     int total)
{
    int g = blockIdx.x * 8 + (threadIdx.x >> 5);
    if (g >= total) return;
    int lane = threadIdx.x & 31;
    int n = g >> 3;
    int h = g & 7;

    float2 w  = *(const float2*)(Wx + (size_t)n * DOUT + h * DP + lane * 2);
    float2 cl = *(const float2*)(aw + h * (2 * DP) + lane * 2);        // tgt coeff
    float2 cr = *(const float2*)(aw + h * (2 * DP) + DP + lane * 2);   // src coeff
    float sr = w.x * cl.x + w.y * cl.y;
    float sl = w.x * cr.x + w.y * cr.y;
#pragma unroll
    for (int off = 16; off; off >>= 1) {
        sr += __shfl_down(sr, off, 32);
        sl += __shfl_down(sl, off, 32);
    }
    if (lane == 0) { er[g] = sr; el[g] = sl; }
}

// ---------------------------------------------------------------------------
// CSR build: deg histogram -> exclusive scan -> cursor copy -> edge-id fill
// ---------------------------------------------------------------------------
__global__ void gat_zero_deg(int* __restrict__ deg, int Nn)
{
    int i = blockIdx.x * blockDim.x + threadIdx.x;
    if (i < Nn) deg[i] = 0;
}

__global__ void gat_hist(const int* __restrict__ ei, int* __restrict__ deg, int E)
{
    int e = blockIdx.x * blockDim.x + threadIdx.x;
    if (e >= E) return;
    atomicAdd(&deg[ei[E + e]], 1);
}

// single-block exclusive scan of deg[0..N) -> offs[0..N], offs[N] = total
__global__ __launch_bounds__(1024) void gat_scan(const int* __restrict__ deg,
                                                 int* __restrict__ offs, int Nn)
{
    __shared__ int buf[1024];
    const int t = threadIdx.x;
    const int chunk = (Nn + 1023) / 1024;
    const int begin = t * chunk;
    const int end   = min(begin + chunk, Nn);

    int sum = 0;
    for (int i = begin; i < end; ++i) sum += deg[i];
    buf[t] = sum;
    __syncthreads();

    // Hillis-Steele inclusive scan over the 1024 chunk sums
    for (int off = 1; off < 1024; off <<= 1) {
        int addend = (t >= off) ? buf[t - off] : 0;
        __syncthreads();
        buf[t] += addend;
        __syncthreads();
    }

    int run = (t == 0) ? 0 : buf[t - 1];     // exclusive base for this chunk
    for (int i = begin; i < end; ++i) {
        offs[i] = run;
        run += deg[i];
    }
    if (t == 0) offs[Nn] = buf[1023];        // total = E
}

__global__ void gat_copy(const int* __restrict__ offs, int* __restrict__ cursor, int Nn)
{
    int i = blockIdx.x * blockDim.x + threadIdx.x;
    if (i < Nn) cursor[i] = offs[i];
}

__global__ void gat_fill(const int* __restrict__ ei, int* __restrict__ cursor,
                         int* __restrict__ elist, int E)
{
    int e = blockIdx.x * blockDim.x + threadIdx.x;
    if (e >= E) return;
    int tgt = ei[E + e];
    int pos = atomicAdd(&cursor[tgt], 1);
    elist[pos] = e;
}

// ---------------------------------------------------------------------------
// Kernel 3: pull-style aggregation with single-pass ONLINE softmax.
// One wave32 per (tgt, head): iterate in-edges, keep running max m, running
// denom d and the 64-wide weighted accumulator (2 floats/lane) in registers.
// No float atomics anywhere; every output element written exactly once.
// ---------------------------------------------------------------------------
__global__ __launch_bounds__(256) void gat_pull(const int* __restrict__ ei,
                                                const int* __restrict__ offs,
                                                const int* __restrict__ elist,
                                                const float* __restrict__ el,
                                                const float* __restrict__ er,
                                                const float* __restrict__ ab,
                                                const float* __restrict__ Wx,
                                                float* __restrict__ out,
                                                int E, int total)
{
    int g = blockIdx.x * 8 + (threadIdx.x >> 5);
    if (g >= total) return;
    int lane = threadIdx.x & 31;
    int n = g >> 3;
    int h = g & 7;

    const int j0 = offs[n];
    const int j1 = offs[n + 1];
    const float base = er[n * NH + h] + ab[h];

    float m = -INFINITY, d = 0.f;
    float2 acc = {0.f, 0.f};

    for (int j = j0; j < j1; ++j) {
        int e   = elist[j];
        int src = ei[e];
        float s = base + el[src * NH + h];
        s = s > 0.f ? s : 0.2f * s;               // leaky_relu(0.2)

        float mn    = fmaxf(m, s);
        float scale = __expf(m - mn);             // 0 on first edge (m = -inf)
        float w     = __expf(s - mn);
        float2 wv = *(const float2*)(Wx + (size_t)src * DOUT + h * DP + lane * 2);
        d     = d * scale + w;
        acc.x = acc.x * scale + w * wv.x;
        acc.y = acc.y * scale + w * wv.y;
        m = mn;
    }

    float inv = d > 0.f ? 1.f / d : 0.f;          // isolated nodes -> zeros
    float2 o = {acc.x * inv, acc.y * inv};
    *(float2*)(out + (size_t)n * DOUT + h * DP + lane * 2) = o;
}

// ---------------------------------------------------------------------------
extern "C" void kernel_launch(void* const* d_in, const int* in_sizes, int n_in,
                              void* d_out, int out_size, void* d_ws, size_t ws_size,
                              hipStream_t stream)
{
    const float* x  = (const float*)d_in[0];
    const int*   ei = (const int*)d_in[1];
    const float* Ws = (const float*)d_in[2];
    const float* Wb = (const float*)d_in[3];
    const float* aw = (const float*)d_in[4];
    const float* ab = (const float*)d_in[5];
    float* out = (float*)d_out;

    const int N = in_sizes[0] / DIN;   // 50000
    const int E = in_sizes[1] / 2;     // 800000

    // workspace layout
    float* Wx     = (float*)d_ws;                        // N*512 f32
    float* el     = Wx + (size_t)N * DOUT;               // N*8
    float* er     = el + (size_t)N * NH;                 // N*8
    int*   deg    = (int*)(er + (size_t)N * NH);         // N
    int*   offs   = deg + N;                             // N+1
    int*   cursor = offs + N + 1;                        // N
    int*   elist  = cursor + N;                          // E

    // dense phase (matrix cores)
    gat_gemm<<<(N + 31) / 32, 256, 0, stream>>>(x, Ws, Wb, Wx, N);
    gat_node_coef<<<(N * NH + 7) / 8, 256, 0, stream>>>(Wx, aw, el, er, N * NH);

    // CSR build over targets
    gat_zero_deg<<<(N + 255) / 256, 256, 0, stream>>>(deg, N);
    gat_hist<<<(E + 255) / 256, 256, 0, stream>>>(ei, deg, E);
    gat_scan<<<1, 1024, 0, stream>>>(deg, offs, N);
    gat_copy<<<(N + 255) / 256, 256, 0, stream>>>(offs, cursor, N);
    gat_fill<<<(E + 255) / 256, 256, 0, stream>>>(ei, cursor, elist, E);

    // pull aggregation with online softmax (no float atomics)
    gat_pull<<<(N * NH + 7) / 8, 256, 0, stream>>>(ei, offs, elist, el, er, ab,
                                                   Wx, out, E, N * NH);
}